// MutualConsistency_68049461838310
// MI455X (gfx1250) — compile-verified
//
#include <hip/hip_runtime.h>
#include <hip/hip_bf16.h>
#include <cfloat>

typedef __attribute__((ext_vector_type(2))) float v2f;
typedef __attribute__((ext_vector_type(8))) float v8f;

#define NB_DICE 1024
#define TPB_DICE 256

// ---------------------------------------------------------------------------
// Kernel 1: fused single-pass reduction over the three masks (bandwidth-bound,
// ~201MB -> ~8.6us at 23.3 TB/s). Emits 5 partial sums per block into d_ws.
// Deterministic (no atomics): two-stage reduction.
// ---------------------------------------------------------------------------
__global__ __launch_bounds__(TPB_DICE)
void dice_partials_kernel(const float* __restrict__ classic,
                          const float* __restrict__ gt,
                          const float* __restrict__ snake,
                          float* __restrict__ partials, int n4) {
    float a0 = 0.f, a1 = 0.f, a2 = 0.f, a3 = 0.f, a4 = 0.f;
    const float4* c4 = (const float4*)classic;
    const float4* g4 = (const float4*)gt;
    const float4* s4 = (const float4*)snake;
    int stride = gridDim.x * blockDim.x;
    for (int i = blockIdx.x * blockDim.x + threadIdx.x; i < n4; i += stride) {
        float4 c = c4[i];
        float4 g = g4[i];
        float4 s = s4[i];
        a0 += c.x + c.y + c.z + c.w;
        a1 += g.x + g.y + g.z + g.w;
        a2 += s.x + s.y + s.z + s.w;
        a3 += c.x * g.x + c.y * g.y + c.z * g.z + c.w * g.w;
        a4 += c.x * s.x + c.y * s.y + c.z * s.z + c.w * s.w;
    }
    __shared__ float red[TPB_DICE];
    float acc[5] = {a0, a1, a2, a3, a4};
    for (int q = 0; q < 5; ++q) {
        red[threadIdx.x] = acc[q];
        __syncthreads();
        for (int off = TPB_DICE / 2; off > 0; off >>= 1) {
            if (threadIdx.x < off) red[threadIdx.x] += red[threadIdx.x + off];
            __syncthreads();
        }
        if (threadIdx.x == 0) partials[q * gridDim.x + blockIdx.x] = red[0];
        __syncthreads();
    }
}

// ---------------------------------------------------------------------------
// Kernel 2: per-(batch,term) circular cross-correlation over all 256 cyclic
// shifts via V_WMMA_F32_16X16X4_F32.  One wave per workgroup; the 16x16 f32
// accumulator tile D[m][n] = corr[16m+n] covers every shift at once:
//   A_wmma[m, k=(j',c)] = a[(j' + 16m) & 255, c]
//   B_wmma[(j',c), n]   = c[(j' -   n) & 255, c]
//   => D[m,n] = sum_j a[j] . c[(j - (16m+n)) & 255]   (K = 512, 128 steps)
// Then loss(s) = (|a|^2 + |c|^2 - 2*corr[s]) / 512, min-reduced in-wave.
// term 0 = seg (A=snake_GT, C=gt_contour, shift0 replaced by snake_classic);
// term 1 = cons (A=snake_classic, C=classic_contour, all shifts).
// ---------------------------------------------------------------------------
__global__ __launch_bounds__(32)
void contour_min_kernel(const float* __restrict__ gt_contour,
                        const float* __restrict__ snake_gt,
                        const float* __restrict__ snake_classic,
                        const float* __restrict__ classic_contour,
                        float* __restrict__ seg_out,
                        float* __restrict__ cons_out) {
    const int b    = blockIdx.x >> 1;
    const int term = blockIdx.x & 1;
    const float* Ap = (term == 0) ? (snake_gt + b * 512) : (snake_classic + b * 512);
    const float* Cp = (term == 0) ? (gt_contour + b * 512) : (classic_contour + b * 512);
    const float* Sp = snake_classic + b * 512;

    __shared__ float lA[512];
    __shared__ float lC[512];
    __shared__ float lS[512];

    const int lane = threadIdx.x;
    #pragma unroll
    for (int i = 0; i < 4; ++i) {
        int idx = lane * 16 + i * 4;
        *(float4*)&lA[idx] = *(const float4*)&Ap[idx];
        *(float4*)&lC[idx] = *(const float4*)&Cp[idx];
        *(float4*)&lS[idx] = *(const float4*)&Sp[idx];
    }
    __syncthreads();

    const int half = lane >> 4;   // which K-pair this lane holds (ISA A 16x4 layout)
    const int lm   = lane & 15;   // M index for A rows / N index for B cols

    v8f acc = {};
    for (int kk = 0; kk < 128; ++kk) {
        // global K indices handled by this lane this step: k0=4kk+2*half, k0+1
        // -> j' = 2kk + half, c = 0 then 1 (consecutive floats in [n,2] layout)
        const int j0 = 2 * kk + half;
        const int ai = ((j0 + 16 * lm) & 255) * 2;   // A[m=lm, k]
        const int bi = ((j0 - lm) & 255) * 2;        // B[k, n=lm]
        v2f af; af.x = lA[ai]; af.y = lA[ai + 1];
        v2f bf; bf.x = lC[bi]; bf.y = lC[bi + 1];
        acc = __builtin_amdgcn_wmma_f32_16x16x4_f32(
            /*neg_a=*/false, af, /*neg_b=*/false, bf,
            /*c_mod=*/(short)0, acc, /*reuse_a=*/false, /*reuse_b=*/false);
    }

    // Scalar norms + the shift-0 special-case dot, each lane covers 16 elems.
    float sa = 0.f, sc = 0.f, ss = 0.f, dsc = 0.f;
    #pragma unroll
    for (int i = 0; i < 16; ++i) {
        float av = lA[lane * 16 + i];
        float cv = lC[lane * 16 + i];
        float sv = lS[lane * 16 + i];
        sa += av * av;
        sc += cv * cv;
        ss += sv * sv;
        dsc += sv * cv;
    }
    #pragma unroll
    for (int off = 16; off > 0; off >>= 1) {
        sa  += __shfl_xor(sa, off, 32);
        sc  += __shfl_xor(sc, off, 32);
        ss  += __shfl_xor(ss, off, 32);
        dsc += __shfl_xor(dsc, off, 32);
    }

    const float inv = 1.0f / 512.0f;
    float mn = FLT_MAX;
    #pragma unroll
    for (int r = 0; r < 8; ++r) {
        // D layout (ISA 7.12.2): VGPR r, lanes 0-15 -> M=r, lanes 16-31 -> M=r+8
        const int s = 16 * ((half == 0) ? r : (r + 8)) + lm;
        float val = (sa + sc - 2.0f * acc[r]) * inv;
        if (term == 0 && s == 0) val = FLT_MAX;   // seg excludes shift 0
        mn = fminf(mn, val);
    }
    #pragma unroll
    for (int off = 16; off > 0; off >>= 1)
        mn = fminf(mn, __shfl_xor(mn, off, 32));

    if (term == 0) {
        // seg shift-0 candidate: mean((snake_classic - gt_contour)^2)
        const float loss0 = (ss + sc - 2.0f * dsc) * inv;
        mn = fminf(mn, loss0);
        if (lane == 0) seg_out[b] = mn;
    } else {
        if (lane == 0) cons_out[b] = mn;
    }
}

// ---------------------------------------------------------------------------
// Kernel 3: final combine (one block). Reduces dice partials deterministically,
// sums the 64 per-batch seg/cons minima, computes the scalar loss.
// ---------------------------------------------------------------------------
__global__ __launch_bounds__(256)
void combine_kernel(const float* __restrict__ partials,
                    const float* __restrict__ segmin,
                    const float* __restrict__ consmin,
                    float* __restrict__ out, int nb) {
    __shared__ float red[256];
    float acc[5];
    for (int q = 0; q < 5; ++q) {
        float s = 0.f;
        for (int i = threadIdx.x; i < nb; i += 256) s += partials[q * nb + i];
        red[threadIdx.x] = s;
        __syncthreads();
        for (int off = 128; off > 0; off >>= 1) {
            if (threadIdx.x < off) red[threadIdx.x] += red[threadIdx.x + off];
            __syncthreads();
        }
        acc[q] = red[0];
        __syncthreads();
    }
    if (threadIdx.x == 0) {
        float segs = 0.f, conss = 0.f;
        for (int i = 0; i < 64; ++i) {
            segs += segmin[i];
            conss += consmin[i];
        }
        const float S_c = acc[0], S_g = acc[1], S_s = acc[2];
        const float I_cg = acc[3], I_cs = acc[4];
        const float dice1 = 1.0f - (2.0f * I_cg + 1.0f) / (S_c + S_g + 1.0f);
        const float dice2 = 1.0f - (2.0f * I_cs + 1.0f) / (S_c + S_s + 1.0f);
        out[0] = dice1 + segs + 0.5f * (dice2 + conss);
    }
}

extern "C" void kernel_launch(void* const* d_in, const int* in_sizes, int n_in,
                              void* d_out, int out_size, void* d_ws, size_t ws_size,
                              hipStream_t stream) {
    // Input order (setup_inputs):
    // 0 ground_truth_mask   [64,512,512] f32
    // 1 ground_truth_contour[64,256,2]   f32
    // 2 snake_GT_size       [64,256,2]   f32
    // 3 snake_classic_size  [64,256,2]   f32
    // 4 snake_mask          [64,512,512] f32
    // 5 classic_contour     [64,256,2]   f32
    // 6 classic_mask        [64,512,512] f32
    const float* gt_mask         = (const float*)d_in[0];
    const float* gt_contour      = (const float*)d_in[1];
    const float* snake_gt        = (const float*)d_in[2];
    const float* snake_classic   = (const float*)d_in[3];
    const float* snake_mask      = (const float*)d_in[4];
    const float* classic_contour = (const float*)d_in[5];
    const float* classic_mask    = (const float*)d_in[6];

    float* ws        = (float*)d_ws;
    float* partials  = ws;                      // 5 * NB_DICE floats
    float* segmin    = ws + 5 * NB_DICE;        // 64 floats
    float* consmin   = ws + 5 * NB_DICE + 64;   // 64 floats
    float* out       = (float*)d_out;

    const int n_mask = in_sizes[0];             // 16,777,216
    const int n4     = n_mask / 4;

    dice_partials_kernel<<<NB_DICE, TPB_DICE, 0, stream>>>(
        classic_mask, gt_mask, snake_mask, partials, n4);

    contour_min_kernel<<<128, 32, 0, stream>>>(
        gt_contour, snake_gt, snake_classic, classic_contour, segmin, consmin);

    combine_kernel<<<1, 256, 0, stream>>>(partials, segmin, consmin, out, NB_DICE);
}